// RMSPELoss_618475290800
// MI455X (gfx1250) — compile-verified
//
#include <hip/hip_runtime.h>
#include <hip/hip_bf16.h>
#include <math.h>

typedef __attribute__((ext_vector_type(2))) float v2f;
typedef __attribute__((ext_vector_type(8))) float v8f;

#define NPERM 720
#define TILES 45            // 720 / 16 perm-rows per WMMA tile
#define KCH   9             // 36 / 4  K-chunks per WMMA
#define WAVES_PER_BLOCK 8
#define THREADS 256

// ---- compile-time lexicographic permutation table (matches itertools.permutations) ----
// each perm packed as 6 x 4-bit digits: sigma[j] = (code >> 4j) & 0xF
struct PermCodes { unsigned c[NPERM]; };
static constexpr PermCodes make_perm_codes() {
    PermCodes t{};
    const int fact[6] = {120, 24, 6, 2, 1, 1};
    for (int p = 0; p < NPERM; ++p) {
        int avail[6] = {0, 1, 2, 3, 4, 5};
        int n = 6, rem = p;
        unsigned code = 0;
        for (int j = 0; j < 6; ++j) {
            int k = rem / fact[j]; rem %= fact[j];
            code |= ((unsigned)avail[k]) << (4 * j);
            for (int q = k; q + 1 < n; ++q) avail[q] = avail[q + 1];
            --n;
        }
        t.c[p] = code;
    }
    return t;
}
__constant__ PermCodes g_perm = make_perm_codes();

__global__ __launch_bounds__(THREADS)
void rmspe_wmma_kernel(const float* __restrict__ pred,
                       const float* __restrict__ doa,
                       float* __restrict__ partial) {
    const int lane = threadIdx.x & 31;
    const int wave = threadIdx.x >> 5;
    const int gw   = blockIdx.x * WAVES_PER_BLOCK + wave;   // global wave id
    const int col  = lane & 15;                              // N within tile
    const bool hi  = lane >= 16;
    const int batch = gw * 16 + col;                         // 16 batches per wave

    // ---- load this lane's row (6 preds + 6 doas), 8B-aligned float2 loads ----
    float pr[6], dv[6];
    {
        const float2* p2 = reinterpret_cast<const float2*>(pred + (size_t)batch * 6);
        const float2* d2 = reinterpret_cast<const float2*>(doa  + (size_t)batch * 6);
        float2 a0 = p2[0], a1 = p2[1], a2 = p2[2];
        float2 b0 = d2[0], b1 = d2[1], b2 = d2[2];
        pr[0] = a0.x; pr[1] = a0.y; pr[2] = a1.x; pr[3] = a1.y; pr[4] = a2.x; pr[5] = a2.y;
        dv[0] = b0.x; dv[1] = b0.y; dv[2] = b1.x; dv[3] = b1.y; dv[4] = b2.x; dv[5] = b2.y;
    }

    const float PI_F   = 3.14159265358979323846f;
    const float HPI    = 1.57079632679489661923f;
    const float INV_PI = 0.31830988618379067154f;

    // ---- 36 pairwise squared wrap-around errors: dl[i*6+j] ----
    float dl[36];
#pragma unroll
    for (int i = 0; i < 6; ++i)
#pragma unroll
        for (int j = 0; j < 6; ++j) {
            float x = pr[i] - dv[j] + HPI;
            float m = x - floorf(x * INV_PI) * PI_F;   // mod(x, pi), result in [0, pi)
            float e = m - HPI;
            dl[i * 6 + j] = e * e;
        }

    // ---- GEMM over perm tiles: S(16 perms x 16 batches) = P_tile(16x36) * d(36x16) ----
    float runmin = 3.402823466e+38f;
    for (int t = 0; t < TILES; ++t) {
        unsigned code = g_perm.c[t * 16 + col];   // this lane's perm row (row = lane%16)
        v8f acc = {0.f, 0.f, 0.f, 0.f, 0.f, 0.f, 0.f, 0.f};
#pragma unroll
        for (int k = 0; k < KCH; ++k) {
            // A fragment: VGPR0 -> K = 4k+{0 lo / 2 hi}, VGPR1 -> K = 4k+{1 lo / 3 hi}
            v2f af, bf;
            {
                const int fl = 4 * k + 0, fh = 4 * k + 2;
                unsigned sh = hi ? (unsigned)(4 * (fh % 6)) : (unsigned)(4 * (fl % 6));
                unsigned iv = hi ? (unsigned)(fh / 6)       : (unsigned)(fl / 6);
                af.x = (((code >> sh) & 0xFu) == iv) ? 1.0f : 0.0f;
            }
            {
                const int fl = 4 * k + 1, fh = 4 * k + 3;
                unsigned sh = hi ? (unsigned)(4 * (fh % 6)) : (unsigned)(4 * (fl % 6));
                unsigned iv = hi ? (unsigned)(fh / 6)       : (unsigned)(fl / 6);
                af.y = (((code >> sh) & 0xFu) == iv) ? 1.0f : 0.0f;
            }
            // B fragment: column = lane%16 (this lane's batch), same K striping
            bf.x = hi ? dl[4 * k + 2] : dl[4 * k + 0];
            bf.y = hi ? dl[4 * k + 3] : dl[4 * k + 1];
            acc = __builtin_amdgcn_wmma_f32_16x16x4_f32(
                false, af, false, bf, (short)0, acc, false, false);
        }
        // running min over the 8 perm-rows this lane holds (other 8 live in lane^16)
        float m0 = fminf(acc[0], acc[1]);
        float m1 = fminf(acc[2], acc[3]);
        float m2 = fminf(acc[4], acc[5]);
        float m3 = fminf(acc[6], acc[7]);
        runmin = fminf(runmin, fminf(fminf(m0, m1), fminf(m2, m3)));
    }

    // combine the two lane-halves that share a batch column
    runmin = fminf(runmin, __shfl_xor(runmin, 16, 32));
    float val = hi ? 0.0f : sqrtf(runmin) * 0.40824829046386301637f;   // 1/sqrt(6)
#pragma unroll
    for (int off = 16; off >= 1; off >>= 1)
        val += __shfl_xor(val, off, 32);
    if (lane == 0) partial[gw] = val;
}

__global__ __launch_bounds__(256)
void rmspe_reduce_kernel(const float* __restrict__ partial, float* __restrict__ out, int n) {
    __shared__ float s[256];
    int t = threadIdx.x;
    float v = 0.0f;
    for (int i = t; i < n; i += 256) v += partial[i];   // fixed order -> deterministic
    s[t] = v;
    __syncthreads();
    for (int off = 128; off > 0; off >>= 1) {
        if (t < off) s[t] += s[t + off];
        __syncthreads();
    }
    if (t == 0) out[0] = s[0];
}

extern "C" void kernel_launch(void* const* d_in, const int* in_sizes, int n_in,
                              void* d_out, int out_size, void* d_ws, size_t ws_size,
                              hipStream_t stream) {
    const float* pred = (const float*)d_in[0];   // doa_predictions (B x 6)
    const float* doa  = (const float*)d_in[1];   // doa             (B x 6)
    float* out = (float*)d_out;
    float* ws  = (float*)d_ws;

    const int nbatch = in_sizes[0] / 6;          // 16384
    const int nwaves = nbatch / 16;              // 1024 partials
    const int blocks = nwaves / WAVES_PER_BLOCK; // 128 blocks of 256 threads

    rmspe_wmma_kernel<<<blocks, THREADS, 0, stream>>>(pred, doa, ws);
    rmspe_reduce_kernel<<<1, 256, 0, stream>>>(ws, out, nwaves);
}